// SystemGraphGNN_86191403696643
// MI455X (gfx1250) — compile-verified
//
#include <hip/hip_runtime.h>
#include <hip/hip_bf16.h>

// Problem constants (match reference)
constexpr int Nn   = 8192;    // nodes
constexpr int En   = 131072;  // edges
constexpr int Bn   = 2;       // batch
constexpr int FN   = 128;     // node feature dim
constexpr int Hd   = 256;     // hidden
constexpr int Ln   = 3;       // conv layers

typedef float v2f __attribute__((ext_vector_type(2)));
typedef float v8f __attribute__((ext_vector_type(8)));

// ---------------------------------------------------------------------------
// Fused GEMM:  C[M,256] = act(A[M,K] @ W[K,256] + bias) (+ addend)
// f32 WMMA 16x16x4. One wave owns a 64-row x 16-col tile (4 accumulators
// sharing each B fragment). Block = 8 waves -> 64 rows x 128 cols.
// The block's K x 128 W tile is staged into LDS with the CDNA5 async-copy
// path (global_load_async_to_lds_b128 / ASYNCcnt); inner-loop B fragments
// come from LDS (ds_load), A fragments stream from global as float2.
// grid = (M/64, 256/128); dynamic LDS = K*128*4 bytes.
// ---------------------------------------------------------------------------
template<int K, int LDA>
__global__ void gemm_wmma_kernel(const float* __restrict__ A,
                                 const float* __restrict__ W,
                                 const float* __restrict__ bias,
                                 float* C,
                                 const float* addend,
                                 int doRelu)
{
    extern __shared__ float sW[];   // [K][128] tile of W
    const int tid  = threadIdx.x;
    const int lane = tid & 31;
    const int wave = tid >> 5;
    const int hl   = lane >> 4;     // half-wave select
    const int l16  = lane & 15;

    const int colbase = blockIdx.y * 128;
    const int wc      = wave * 16 + l16;   // column within the 128-wide tile
    const int col     = colbase + wc;
    const int rbase   = blockIdx.x * 64;

    // ---- async-stage W[K x 128] into LDS (ASYNCcnt path) ----
    {
        const unsigned ldsBase = (unsigned)(uintptr_t)sW;
        constexpr int NV4 = K * 128 / 4;          // float4 tiles to move
        for (int idx = tid; idx < NV4; idx += 256) {
            const int row = idx >> 5;             // 32 float4 per 128-col row
            const int c4  = idx & 31;
            const float* gp  = W + (size_t)row * Hd + colbase + c4 * 4;
            const unsigned lp = ldsBase + (unsigned)((row * 128 + c4 * 4) * 4);
            asm volatile("global_load_async_to_lds_b128 %0, %1, off"
                         :: "v"(lp), "v"((unsigned long long)(uintptr_t)gp)
                         : "memory");
        }
        asm volatile("s_wait_asynccnt 0x0" ::: "memory");
        __syncthreads();
    }

    v8f acc[4] = {};

#pragma unroll 4
    for (int k = 0; k < K; k += 4) {
        // B fragment (4x16): VGPR j holds rows k + 2*hl + j, column l16
        v2f b;
        b.x = sW[(k + 2 * hl)     * 128 + wc];
        b.y = sW[(k + 2 * hl + 1) * 128 + wc];
#pragma unroll
        for (int t = 0; t < 4; ++t) {
            // A fragment (16x4): lane holds A[row, k+2*hl .. k+2*hl+1]
            const float* ap = A + (size_t)(rbase + t * 16 + l16) * LDA + k + 2 * hl;
            v2f a = *(const v2f*)ap;
            acc[t] = __builtin_amdgcn_wmma_f32_16x16x4_f32(
                false, a, false, b, (short)0, acc[t], false, false);
        }
    }

    const float bv = bias[col];
#pragma unroll
    for (int t = 0; t < 4; ++t) {
#pragma unroll
        for (int r = 0; r < 8; ++r) {
            // C/D layout: VGPR r -> row r (lanes 0-15) / row r+8 (lanes 16-31)
            const int row = rbase + t * 16 + r + 8 * hl;
            float v = acc[t][r] + bv;
            if (doRelu) v = fmaxf(v, 0.0f);
            if (addend) v += addend[(size_t)row * Hd + col];
            C[(size_t)row * Hd + col] = v;
        }
    }
}

// ---------------------------------------------------------------------------
// Sparse adjacency apply: m[b,i,:] = sum over edges (src==i) of h[b,dst,:]
// src is sorted, so edges of node i form a contiguous range -> binary search.
// grid = N blocks, block = H threads (one per channel), both batches per block.
// ---------------------------------------------------------------------------
__device__ __forceinline__ int lower_bound_i(const int* __restrict__ a, int n, int key)
{
    int lo = 0, hi = n;
    while (lo < hi) {
        int mid = (lo + hi) >> 1;
        if (a[mid] < key) lo = mid + 1; else hi = mid;
    }
    return lo;
}

__global__ void spmm_kernel(const float* __restrict__ h,
                            const int* __restrict__ src,
                            const int* __restrict__ dst,
                            float* __restrict__ m)
{
    const int i   = blockIdx.x;
    const int tid = threadIdx.x;
    const int lo  = lower_bound_i(src, En, i);
    const int hi  = lower_bound_i(src, En, i + 1);

    float a0 = 0.0f, a1 = 0.0f;
    for (int e = lo; e < hi; ++e) {
        const int d = dst[e];
        if (e + 1 < hi) {
            const int dn = dst[e + 1];     // pull next gather row toward WGP
            __builtin_prefetch(&h[(size_t)dn * Hd + tid], 0, 0);
            __builtin_prefetch(&h[(size_t)(Nn + dn) * Hd + tid], 0, 0);
        }
        a0 += h[(size_t)d * Hd + tid];                 // batch 0
        a1 += h[(size_t)(Nn + d) * Hd + tid];          // batch 1
    }
    m[(size_t)i * Hd + tid]        = a0;
    m[(size_t)(Nn + i) * Hd + tid] = a1;
}

// ---------------------------------------------------------------------------
// Per-node heads: 5 classifier dots, the two W_ep half-dots (p1/p2), and the
// W_gp dot (accumulated for the global mean). grid = B*N, block = H.
// ---------------------------------------------------------------------------
__global__ void heads_kernel(const float* __restrict__ h,
                             const float* __restrict__ Wcls,
                             const float* __restrict__ bcls,
                             const float* __restrict__ Wep,
                             const float* __restrict__ Wgp,
                             float* __restrict__ node_health,
                             float* __restrict__ p1,
                             float* __restrict__ p2,
                             float* __restrict__ gacc)
{
    __shared__ float sh[Hd];
    const int row = blockIdx.x;     // b*N + i
    const int tid = threadIdx.x;
    const float x = h[(size_t)row * Hd + tid];

    float w[8];
#pragma unroll
    for (int c = 0; c < 5; ++c) w[c] = Wcls[tid * 5 + c];
    w[5] = Wep[tid];
    w[6] = Wep[Hd + tid];
    w[7] = Wgp[tid];

    for (int o = 0; o < 8; ++o) {
        sh[tid] = x * w[o];
        __syncthreads();
        for (int s = Hd / 2; s > 0; s >>= 1) {
            if (tid < s) sh[tid] += sh[tid + s];
            __syncthreads();
        }
        if (tid == 0) {
            const float r = sh[0];
            if (o < 5)       node_health[(size_t)row * 5 + o] = r + bcls[o];
            else if (o == 5) p1[row] = r;
            else if (o == 6) p2[row] = r;
            else             atomicAdd(&gacc[row / Nn], r);
        }
        __syncthreads();
    }
}

// ---------------------------------------------------------------------------
// edge_propagation[b,e] = sigmoid(p1[b,src[e]] + p2[b,dst[e]] + b_ep)
// ---------------------------------------------------------------------------
__global__ void edge_kernel(const float* __restrict__ p1,
                            const float* __restrict__ p2,
                            const int* __restrict__ src,
                            const int* __restrict__ dst,
                            const float* __restrict__ bep,
                            float* __restrict__ out)
{
    const int idx = blockIdx.x * blockDim.x + threadIdx.x;
    if (idx >= Bn * En) return;
    const int b = idx / En;
    const int e = idx - b * En;
    const float z = p1[b * Nn + src[e]] + p2[b * Nn + dst[e]] + bep[0];
    out[idx] = 1.0f / (1.0f + __expf(-z));
}

__global__ void finalize_kernel(const float* __restrict__ gacc,
                                const float* __restrict__ bgp,
                                float* __restrict__ out_sys)
{
    const int b = threadIdx.x;
    if (b < Bn) {
        const float z = gacc[b] * (1.0f / (float)Nn) + bgp[0];
        out_sys[b] = 1.0f / (1.0f + __expf(-z));
    }
}

// ---------------------------------------------------------------------------
extern "C" void kernel_launch(void* const* d_in, const int* in_sizes, int n_in,
                              void* d_out, int out_size, void* d_ws, size_t ws_size,
                              hipStream_t stream)
{
    (void)in_sizes; (void)n_in; (void)out_size; (void)ws_size;

    const float* node_features = (const float*)d_in[0];
    // d_in[1] edge_features : unused (dead code in reference)
    // d_in[2] adjacency     : unused (sparse path via src/dst)
    const int*   src    = (const int*)d_in[3];
    const int*   dst    = (const int*)d_in[4];
    const float* W_ne   = (const float*)d_in[5];
    const float* b_ne   = (const float*)d_in[6];
    // d_in[7] W_ee, d_in[8] b_ee : unused
    const float* W_conv = (const float*)d_in[9];
    const float* b_conv = (const float*)d_in[10];
    const float* W_cls  = (const float*)d_in[11];
    const float* b_cls  = (const float*)d_in[12];
    const float* W_ep   = (const float*)d_in[13];
    const float* b_ep   = (const float*)d_in[14];
    const float* W_gp   = (const float*)d_in[15];
    const float* b_gp   = (const float*)d_in[16];

    // workspace layout (f32)
    float* ws   = (float*)d_ws;
    float* h    = ws;                                  // B*N*H
    float* m    = h  + (size_t)Bn * Nn * Hd;           // B*N*H
    float* p1   = m  + (size_t)Bn * Nn * Hd;           // B*N
    float* p2   = p1 + (size_t)Bn * Nn;                // B*N
    float* gacc = p2 + (size_t)Bn * Nn;                // B

    // output layout: node_health | edge_propagation | system_health | h
    float* out     = (float*)d_out;
    float* out_nh  = out;                              // B*N*5
    float* out_ep  = out_nh  + (size_t)Bn * Nn * 5;    // B*E
    float* out_sys = out_ep  + (size_t)Bn * En;        // B
    float* out_h   = out_sys + Bn;                     // B*N*H

    hipMemsetAsync(gacc, 0, Bn * sizeof(float), stream);

    const int M = Bn * Nn;  // 16384 rows

    // h = relu(X @ W_ne + b_ne)   (K=128 -> 64KB LDS W tile)
    gemm_wmma_kernel<FN, FN><<<dim3(M / 64, Hd / 128), 256,
                               FN * 128 * sizeof(float), stream>>>(
        node_features, W_ne, b_ne, h, nullptr, 1);

    // L rounds of: m = A_sparse @ h ; h = relu(m @ W_conv[l] + b_conv[l]) + h
    for (int l = 0; l < Ln; ++l) {
        spmm_kernel<<<Nn, Hd, 0, stream>>>(h, src, dst, m);
        gemm_wmma_kernel<Hd, Hd><<<dim3(M / 64, Hd / 128), 256,
                                   Hd * 128 * sizeof(float), stream>>>(
            m, W_conv + (size_t)l * Hd * Hd, b_conv + l * Hd, h, h, 1);
    }

    // heads
    heads_kernel<<<M, Hd, 0, stream>>>(h, W_cls, b_cls, W_ep, W_gp,
                                       out_nh, p1, p2, gacc);
    edge_kernel<<<(Bn * En + 255) / 256, 256, 0, stream>>>(p1, p2, src, dst,
                                                           b_ep, out_ep);
    finalize_kernel<<<1, 32, 0, stream>>>(gacc, b_gp, out_sys);

    // h is itself an output
    hipMemcpyAsync(out_h, h, (size_t)Bn * Nn * Hd * sizeof(float),
                   hipMemcpyDeviceToDevice, stream);
}